// GraphConv_Custom_14267881357535
// MI455X (gfx1250) — compile-verified
//
#include <hip/hip_runtime.h>

typedef __attribute__((ext_vector_type(2))) float v2f;
typedef __attribute__((ext_vector_type(8))) float v8f;

#define N_NODES 100000
#define DFEAT   256
#define MTILES  (N_NODES / 16)   // 6250 row tiles of 16

// ---------------------------------------------------------------------------
// 1) Zero workspace: agg accumulator + degree counters (must run every call)
// ---------------------------------------------------------------------------
__global__ void zero_ws_kernel(float* __restrict__ agg,
                               float* __restrict__ odeg,
                               float* __restrict__ ideg) {
    size_t i = (size_t)blockIdx.x * blockDim.x + threadIdx.x;
    const size_t total = (size_t)N_NODES * DFEAT;
    if (i < total) agg[i] = 0.0f;
    if (i < (size_t)N_NODES) { odeg[i] = 0.0f; ideg[i] = 0.0f; }
}

// ---------------------------------------------------------------------------
// 2) Degree histograms via non-returning global float atomics
// ---------------------------------------------------------------------------
__global__ void degree_kernel(const int* __restrict__ src,
                              const int* __restrict__ dst,
                              int E,
                              float* __restrict__ odeg,
                              float* __restrict__ ideg) {
    int e = blockIdx.x * blockDim.x + threadIdx.x;
    if (e < E) {
        atomicAdd(&odeg[src[e]], 1.0f);
        atomicAdd(&ideg[dst[e]], 1.0f);
    }
}

// ---------------------------------------------------------------------------
// 3) deg -> clip(deg,1)^-0.5 in place
// ---------------------------------------------------------------------------
__global__ void invsqrt_deg_kernel(float* __restrict__ odeg,
                                   float* __restrict__ ideg) {
    int i = blockIdx.x * blockDim.x + threadIdx.x;
    if (i < N_NODES) {
        odeg[i] = rsqrtf(fmaxf(odeg[i], 1.0f));
        ideg[i] = rsqrtf(fmaxf(ideg[i], 1.0f));
    }
}

// ---------------------------------------------------------------------------
// 4) Edge-parallel SpMM with fused pre-scale:
//    agg[dst[e], :] += feat[src[e], :] * odeg_inv[src[e]]
//    64 threads per edge, float4 gather, 4 atomics per thread (L2-resident).
// ---------------------------------------------------------------------------
__global__ void spmm_kernel(const float* __restrict__ feat,
                            const int* __restrict__ src,
                            const int* __restrict__ dst,
                            const float* __restrict__ oinv,
                            float* __restrict__ agg,
                            int E) {
    long tid = (long)blockIdx.x * blockDim.x + threadIdx.x;
    int e    = (int)(tid >> 6);   // 64 threads per edge
    int lane = (int)(tid & 63);   // each thread: 4 consecutive floats
    if (e >= E) return;
    int s = src[e];
    int d = dst[e];
    float sc = oinv[s];
    const float4 v = *reinterpret_cast<const float4*>(feat + (size_t)s * DFEAT + lane * 4);
    float* o = agg + (size_t)d * DFEAT + lane * 4;
    atomicAdd(o + 0, v.x * sc);
    atomicAdd(o + 1, v.y * sc);
    atomicAdd(o + 2, v.z * sc);
    atomicAdd(o + 3, v.w * sc);
}

// ---------------------------------------------------------------------------
// 5) Dense GEMM: out = (agg @ W) * ideg_inv[:,None] + bias, fp32 WMMA.
//    One wave per 16x64 output strip: 4 accumulators, A reused 4x per K-step.
//    VGPR layouts per CDNA5 ISA 7.12.2:
//      A(16x4):  lanes 0-15 -> K={k,k+1}; lanes 16-31 -> K={k+2,k+3}
//      B(4x16):  VGPR0 = rows {k (lanes 0-15), k+2 (lanes 16-31)}, VGPR1 = {k+1, k+3}
//      C/D(16x16): VGPR j = rows {j (lanes 0-15), j+8 (lanes 16-31)}
// ---------------------------------------------------------------------------
__global__ void gemm_wmma_kernel(const float* __restrict__ agg,
                                 const float* __restrict__ W,
                                 const float* __restrict__ bias,
                                 const float* __restrict__ iinv,
                                 float* __restrict__ out) {
    int wave = blockIdx.x * (blockDim.x >> 5) + (threadIdx.x >> 5);
    int lane = threadIdx.x & 31;
    int tileM = wave >> 2;       // 16-row tile
    int ngrp  = wave & 3;        // 64-column group
    if (tileM >= MTILES) return;

    const int half = lane >> 4;  // 0: lanes 0-15, 1: lanes 16-31
    const int l15  = lane & 15;

    const float* arow = agg + (size_t)(tileM * 16 + l15) * DFEAT;
    const int colbase = ngrp * 64;
    const int ncol    = colbase + l15;

    v8f acc0 = {}, acc1 = {}, acc2 = {}, acc3 = {};

    #pragma unroll 4
    for (int k = 0; k < DFEAT; k += 4) {
        const int ka = k + half * 2;
        v2f a;
        a.x = arow[ka];
        a.y = arow[ka + 1];

        const float* b0 = W + (size_t)ka * DFEAT + ncol;        // row ka
        const float* b1 = b0 + DFEAT;                           // row ka+1
        v2f bb0, bb1, bb2, bb3;
        bb0.x = b0[0];   bb0.y = b1[0];
        bb1.x = b0[16];  bb1.y = b1[16];
        bb2.x = b0[32];  bb2.y = b1[32];
        bb3.x = b0[48];  bb3.y = b1[48];

        acc0 = __builtin_amdgcn_wmma_f32_16x16x4_f32(false, a, false, bb0, (short)0, acc0, false, false);
        acc1 = __builtin_amdgcn_wmma_f32_16x16x4_f32(false, a, false, bb1, (short)0, acc1, false, false);
        acc2 = __builtin_amdgcn_wmma_f32_16x16x4_f32(false, a, false, bb2, (short)0, acc2, false, false);
        acc3 = __builtin_amdgcn_wmma_f32_16x16x4_f32(false, a, false, bb3, (short)0, acc3, false, false);
    }

    // Fused post-scale + bias store
    #pragma unroll
    for (int j = 0; j < 8; ++j) {
        const int row = tileM * 16 + j + 8 * half;
        const float p = iinv[row];
        float* orow = out + (size_t)row * DFEAT;
        const int c0 = colbase + l15;
        orow[c0 +  0] = acc0[j] * p + bias[c0 +  0];
        orow[c0 + 16] = acc1[j] * p + bias[c0 + 16];
        orow[c0 + 32] = acc2[j] * p + bias[c0 + 32];
        orow[c0 + 48] = acc3[j] * p + bias[c0 + 48];
    }
}

// ---------------------------------------------------------------------------
// Launcher
// ---------------------------------------------------------------------------
extern "C" void kernel_launch(void* const* d_in, const int* in_sizes, int n_in,
                              void* d_out, int out_size, void* d_ws, size_t ws_size,
                              hipStream_t stream) {
    const float* feat = (const float*)d_in[0];
    const int*   src  = (const int*)d_in[1];
    const int*   dst  = (const int*)d_in[2];
    const float* W    = (const float*)d_in[3];
    const float* bias = (const float*)d_in[4];
    const int E = in_sizes[1];
    float* out = (float*)d_out;

    // workspace layout: agg [N*D] | odeg [N] | ideg [N]
    float* agg  = (float*)d_ws;
    float* odeg = agg + (size_t)N_NODES * DFEAT;
    float* ideg = odeg + N_NODES;

    const size_t total = (size_t)N_NODES * DFEAT;
    {
        int blocks = (int)((total + 255) / 256);
        zero_ws_kernel<<<blocks, 256, 0, stream>>>(agg, odeg, ideg);
    }
    {
        int blocks = (E + 255) / 256;
        degree_kernel<<<blocks, 256, 0, stream>>>(src, dst, E, odeg, ideg);
    }
    {
        int blocks = (N_NODES + 255) / 256;
        invsqrt_deg_kernel<<<blocks, 256, 0, stream>>>(odeg, ideg);
    }
    {
        long threads = (long)E * 64;
        long blocks = (threads + 255) / 256;
        spmm_kernel<<<(int)blocks, 256, 0, stream>>>(feat, src, dst, odeg, agg, E);
    }
    {
        // one wave (32 threads) per 16x64 output strip; 8 waves per block
        int waves = MTILES * 4;
        int blocks = (waves + 7) / 8;
        gemm_wmma_kernel<<<blocks, 256, 0, stream>>>(agg, W, bias, ideg, out);
    }
}